// ChamferLoss_47141561041505
// MI455X (gfx1250) — compile-verified
//
#include <hip/hip_runtime.h>

typedef __attribute__((ext_vector_type(2))) float v2f;
typedef __attribute__((ext_vector_type(8))) float v8f;

#define BATCH 8
#define NDIM  3

// One wave per (batch, 16-row tile of "rows" points); 4 waves per block.
// For each row point, computes min over all "cols" points of squared distance.
// The entire distance formula is folded into one V_WMMA_F32_16X16X4_F32:
//   A row  = (-2x, -2y, -2z, 1)        (K=3 padded, -2 pre-applied)
//   B col  = (gx,  gy,  gz,  |g|^2)    (norm rides in the K3 slot)
//   C      = |p|^2 per output row
//   => D[m][n] = |p_m|^2 + |g_n|^2 - 2<p_m, g_n> = d2[m][n]
// sqrt/max(.,0) commute with min (monotone), so sqrt happens once per row.
__global__ __launch_bounds__(128) void chamfer_rowmin_wmma(
    const float* __restrict__ rows,   // (B,3,N)
    const float* __restrict__ cols,   // (B,3,N)
    float* __restrict__ rowmin_out,   // (B,N)
    int N)
{
  const int  lane = threadIdx.x;   // 0..31
  const int  m16  = lane & 15;
  const bool hi   = lane >= 16;
  const int  b    = blockIdx.y;
  const int  row0 = (blockIdx.x * 4 + threadIdx.y) * 16;

  const float* rp = rows + (size_t)b * NDIM * N;
  const float* cp = cols + (size_t)b * NDIM * N;

  // Per-lane component base: low lanes read x (K0), hi lanes read z (K2).
  const int base0 = hi ? 2 * N : 0;

  // ---- A tile (16x4 f32): lanes 0-15 {VGPR0:-2x, VGPR1:-2y},
  //                         lanes 16-31 {VGPR0:-2z, VGPR1: 1}.
  const int ar = row0 + m16;
  float rx = rp[base0 + ar];          // low: x, hi: z
  float ry = rp[N + ar];              // y (valid for all lanes)
  float ry0 = hi ? 0.0f : ry;
  // |p|^2 replicated: low half contributes x^2+y^2, high half z^2.
  float t   = fmaf(ry0, ry0, rx * rx);
  float pn2 = t + __shfl_xor(t, 16);
  v2f A;
  A.x = -2.0f * rx;
  A.y = hi ? 1.0f : -2.0f * ry;

  // Output VGPR r holds row M = r + 8*hi -> that row's norm is on lane r(+8).
  v8f cinit;
#pragma unroll
  for (int r = 0; r < 8; ++r)
    cinit[r] = __shfl(pn2, r + (hi ? 8 : 0));

  float rm[8];
#pragma unroll
  for (int r = 0; r < 8; ++r) rm[r] = 3.0e38f;

  for (int col0 = 0; col0 < N; col0 += 16) {
    const int bc = col0 + m16;
    float gx = cp[base0 + bc];        // low: gx, hi: gz  (branchless, no exec mask)
    float gy = cp[N + bc];            // unconditional load, valid everywhere
    float gy0 = hi ? 0.0f : gy;
    float s   = fmaf(gy0, gy0, gx * gx);     // low: gx^2+gy^2, hi: gz^2
    float gn2 = s + __shfl_xor(s, 16);       // |g|^2 on all lanes

    v2f Bm;
    Bm.x = gx;
    Bm.y = hi ? gn2 : gy;             // K3 slot carries |g|^2

    // d2 tile straight out of the matrix pipe.
    v8f c = __builtin_amdgcn_wmma_f32_16x16x4_f32(
        false, A, false, Bm, (short)0, cinit, false, false);

#pragma unroll
    for (int r = 0; r < 8; ++r)
      rm[r] = fminf(rm[r], c[r]);
  }

  // Min across each 16-lane group (lanes 0-15: rows r, lanes 16-31: rows r+8).
#pragma unroll
  for (int r = 0; r < 8; ++r) {
    float v = rm[r];
    v = fminf(v, __shfl_xor(v, 1));
    v = fminf(v, __shfl_xor(v, 2));
    v = fminf(v, __shfl_xor(v, 4));
    v = fminf(v, __shfl_xor(v, 8));
    rm[r] = v;
  }

  float* out = rowmin_out + (size_t)b * N + row0;
  if (lane == 0) {
#pragma unroll
    for (int r = 0; r < 8; ++r) out[r] = sqrtf(fmaxf(rm[r], 0.0f));
  } else if (lane == 16) {
#pragma unroll
    for (int r = 0; r < 8; ++r) out[8 + r] = sqrtf(fmaxf(rm[r], 0.0f));
  }
}

// Deterministic single-block reduction of both min arrays -> scalar loss.
__global__ __launch_bounds__(256) void chamfer_reduce(
    const float* __restrict__ mins, float* __restrict__ out,
    int total, float inv_denom)
{
  __shared__ float sm[256];
  float s = 0.0f;
  for (int i = threadIdx.x; i < total; i += 256) s += mins[i];
  sm[threadIdx.x] = s;
  __syncthreads();
  for (int off = 128; off > 0; off >>= 1) {
    if ((int)threadIdx.x < off) sm[threadIdx.x] += sm[threadIdx.x + off];
    __syncthreads();
  }
  if (threadIdx.x == 0) out[0] = sm[0] * inv_denom;
}

extern "C" void kernel_launch(void* const* d_in, const int* in_sizes, int n_in,
                              void* d_out, int out_size, void* d_ws, size_t ws_size,
                              hipStream_t stream) {
  const float* pred = (const float*)d_in[0];  // (8,3,4096) f32
  const float* gt   = (const float*)d_in[1];  // (8,3,4096) f32

  const int N = in_sizes[0] / (BATCH * NDIM); // 4096

  float* ws_z2 = (float*)d_ws;                 // (B,N): min over gt per pred pt
  float* ws_z1 = ws_z2 + (size_t)BATCH * N;    // (B,N): min over pred per gt pt

  dim3 block(32, 4);
  dim3 grid(N / 64, BATCH);
  chamfer_rowmin_wmma<<<grid, block, 0, stream>>>(pred, gt, ws_z2, N);
  chamfer_rowmin_wmma<<<grid, block, 0, stream>>>(gt, pred, ws_z1, N);

  const int total = 2 * BATCH * N;
  chamfer_reduce<<<1, 256, 0, stream>>>(ws_z2, (float*)d_out, total,
                                        1.0f / (float)(BATCH * N));
}